// Warp_Tensors_Affine_Layer_76020921139526
// MI455X (gfx1250) — compile-verified
//
#include <hip/hip_runtime.h>
#include <math.h>

// Problem constants from the reference: (B, C, H, W) = (4, 3, 768, 768), fp32.
#define NB 4
#define NC 3
#define NH 768
#define NW 768

typedef float v2f __attribute__((ext_vector_type(2)));
typedef float v8f __attribute__((ext_vector_type(8)));

// 16-byte row fetch with only 4-byte alignment guarantee (taps start at x0-1).
struct __attribute__((packed)) F4 { float a, b, c, d; };

#if __has_builtin(__builtin_amdgcn_wmma_f32_16x16x4_f32)
#define USE_WMMA_COEFFS 1
#else
#define USE_WMMA_COEFFS 0
#endif

// Catmull-Rom coefficients for one per-lane t, for all 32 lanes of a wave.
// WMMA formulation: c = M (4x4, shared) x T (4x16, per-pixel powers).
//   A rows 0-3 = M and rows 8-11 = M  ->  D rows 8-11 (which live in lanes
//   16-31) replicate the result, so batch0 (pixels=lanes 0-15) is taken from
//   the low half of D0 and batch1 (pixels=lanes 16-31) from the high half of
//   D1 with plain v_cndmask merges - no cross-lane moves on the output side.
// B layout (K=4, 2 VGPRs): v0 = {row0 | row2}, v1 = {row1 | row3} across the
// two lane halves, consistent with the documented A / sparse-B striping.
__device__ __forceinline__ void cubic_coeffs_wave(
    float t, bool hi, v2f A, float& c0, float& c1, float& c2, float& c3)
{
#if USE_WMMA_COEFFS
  const float tsw = __shfl_xor(t, 16, 32);
  // batch0 needs t of pixel (lane&15) in BOTH lane halves; batch1 the t of
  // pixel 16+(lane&15).
  const float tb0 = hi ? tsw : t;
  const float tb1 = hi ? t : tsw;

  const float p0_2 = tb0 * tb0, p0_3 = p0_2 * tb0;
  const float p1_2 = tb1 * tb1, p1_3 = p1_2 * tb1;

  v2f B0, B1;
  B0.x = hi ? tb0 : p0_3;   // low: row0 = t^3, high: row2 = t
  B0.y = hi ? 1.0f : p0_2;  // low: row1 = t^2, high: row3 = 1
  B1.x = hi ? tb1 : p1_3;
  B1.y = hi ? 1.0f : p1_2;

  v8f z = {};
  v8f d0 = __builtin_amdgcn_wmma_f32_16x16x4_f32(
      false, A, false, B0, (short)0, z, false, false);
  v8f d1 = __builtin_amdgcn_wmma_f32_16x16x4_f32(
      false, A, false, B1, (short)0, z, false, false);

  c0 = hi ? d1[0] : d0[0];
  c1 = hi ? d1[1] : d0[1];
  c2 = hi ? d1[2] : d0[2];
  c3 = hi ? d1[3] : d0[3];
#else
  (void)hi; (void)A;
  const float t2 = t * t, t3 = t2 * t;
  c0 = 0.5f * (-t3 + 2.0f * t2 - t);
  c1 = 0.5f * (3.0f * t3 - 5.0f * t2 + 2.0f);
  c2 = 0.5f * (-3.0f * t3 + 4.0f * t2 + t);
  c3 = 0.5f * (t3 - t2);
#endif
}

__global__ __launch_bounds__(256) void warp_bicubic_kernel(
    const float* __restrict__ img,
    const float* __restrict__ shift_x,
    const float* __restrict__ shift_y,
    const float* __restrict__ scale,
    const float* __restrict__ rot_deg,
    float* __restrict__ out)
{
  // ---- Build the shared A operand (Catmull-Rom matrix) once per thread ----
  const int lane = (int)(threadIdx.x & 31u);
  const bool hi = lane >= 16;
  const int r = lane & 15;          // D/A row this lane slot feeds
  const int m = r & 3;              // M row index (rows 0-3 and 8-11 hold M)
  const bool nz = (r & 4) == 0;
  // M columns in power order [t^3, t^2, t, 1]:
  //   cm1 = -.5t^3 +  t^2 - .5t      c0 = 1.5t^3 - 2.5t^2 + 1
  //   c1  = -1.5t^3 + 2t^2 + .5t     c2 =  .5t^3 -  .5t^2
  const float col0 = (m == 0) ? -0.5f : (m == 1) ? 1.5f : (m == 2) ? -1.5f : 0.5f;
  const float col1 = (m == 0) ?  1.0f : (m == 1) ? -2.5f : (m == 2) ?  2.0f : -0.5f;
  const float col2 = (m == 0) ? -0.5f : (m == 2) ? 0.5f : 0.0f;
  const float col3 = (m == 1) ?  1.0f : 0.0f;
  v2f A;
  A.x = nz ? (hi ? col2 : col0) : 0.0f;  // K = hi?2:0
  A.y = nz ? (hi ? col3 : col1) : 0.0f;  // K = hi?3:1

  // ---- Per-pixel affine inverse map (params uniform per block -> SMEM) ----
  const int bc = (int)blockIdx.z;        // 0 .. B*C-1
  const int b  = bc / NC;
  const size_t plane = (size_t)bc * (size_t)(NH * NW);

  const int p = (int)(blockIdx.x * blockDim.x + threadIdx.x);  // 0 .. H*W-1
  const int px = p % NW;
  const int py = p / NW;

  const float sx = shift_x[b], sy = shift_y[b];
  const float sc = scale[b];
  const float sx0 = shift_x[0], sy0 = shift_y[0], sc0 = scale[0];

  // Centering uses batch-0 corner values exactly as the reference does.
  const float cxc = (((float)(NW - 1) + sx0) / sc0 - sx0 / sc0) * 0.5f;
  const float cyc = (((float)(NH - 1) + sy0) / sc0 - sy0 / sc0) * 0.5f;

  const float th = rot_deg[b] * 0.017453292519943295f;  // pi/180
  const float ct = __cosf(th), st = __sinf(th);

  const float X1 = ((float)px + sx) / sc;
  const float Y1 = ((float)py + sy) / sc;
  const float Xc = X1 - cxc, Yc = Y1 - cyc;
  const float Xn = ct * Xc - st * Yc;
  const float Yn = st * Xc + ct * Yc;

  const float halfW = (float)(NW - 1) * 0.5f;
  const float halfH = (float)(NH - 1) * 0.5f;
  // Mirror the reference's normalize->denormalize round trip bit-for-bit-ish.
  const float xm = (Xn / halfW + 1.0f) * halfW;
  const float ym = (Yn / halfH + 1.0f) * halfH;

  const float x0f = floorf(xm), y0f = floorf(ym);
  const float tx = xm - x0f,  ty = ym - y0f;
  const int x0 = (int)x0f, y0 = (int)y0f;

  // ---- Cubic coefficients via WMMA (2 x V_WMMA_F32_16X16X4_F32 per axis).
  // EXEC is all-1s here: full blocks, no divergence has occurred yet.
  float cy0, cy1, cy2, cy3, cx0, cx1, cx2, cx3;
  cubic_coeffs_wave(ty, hi, A, cy0, cy1, cy2, cy3);
  cubic_coeffs_wave(tx, hi, A, cx0, cx1, cx2, cx3);

  // ---- 16-tap gather: interior fast path = 4 unaligned b128 row loads ----
  const float* __restrict__ base = img + plane;
  float acc = 0.0f;

  if (x0 >= 1 && x0 <= NW - 3 && y0 >= 1 && y0 <= NH - 3) {
    const float* rp = base + (size_t)(y0 - 1) * NW + (x0 - 1);
#pragma unroll
    for (int i = 0; i < 4; ++i) {
      const F4 v = *reinterpret_cast<const F4*>(rp);
      float d = v.a * cx0;
      d = fmaf(v.b, cx1, d);
      d = fmaf(v.c, cx2, d);
      d = fmaf(v.d, cx3, d);
      const float cyi = (i == 0) ? cy0 : (i == 1) ? cy1 : (i == 2) ? cy2 : cy3;
      acc = fmaf(cyi, d, acc);
      rp += NW;
    }
  } else {
    // Border: per-tap clamp (replicate edge), matching jnp.clip in reference.
    int xs[4], ys[4];
#pragma unroll
    for (int j = 0; j < 4; ++j) {
      int xx = x0 - 1 + j;
      xs[j] = xx < 0 ? 0 : (xx > NW - 1 ? NW - 1 : xx);
      int yy = y0 - 1 + j;
      ys[j] = yy < 0 ? 0 : (yy > NH - 1 ? NH - 1 : yy);
    }
#pragma unroll
    for (int i = 0; i < 4; ++i) {
      const float* rowp = base + (size_t)ys[i] * NW;
      float d = rowp[xs[0]] * cx0;
      d = fmaf(rowp[xs[1]], cx1, d);
      d = fmaf(rowp[xs[2]], cx2, d);
      d = fmaf(rowp[xs[3]], cx3, d);
      const float cyi = (i == 0) ? cy0 : (i == 1) ? cy1 : (i == 2) ? cy2 : cy3;
      acc = fmaf(cyi, d, acc);
    }
  }

  out[plane + (size_t)p] = acc;  // coalesced b32 store, 128B per wave
}

extern "C" void kernel_launch(void* const* d_in, const int* in_sizes, int n_in,
                              void* d_out, int out_size, void* d_ws, size_t ws_size,
                              hipStream_t stream) {
  (void)in_sizes; (void)n_in; (void)out_size; (void)d_ws; (void)ws_size;
  const float* img = (const float*)d_in[0];
  const float* sx  = (const float*)d_in[1];
  const float* sy  = (const float*)d_in[2];
  const float* sc  = (const float*)d_in[3];
  const float* rot = (const float*)d_in[4];
  float* out = (float*)d_out;

  // H*W = 589824 pixels per (b,c) plane; 256-thread blocks (8 wave32) divide
  // it exactly, so every wave is full (EXEC all-1s, as WMMA requires).
  dim3 grid((NH * NW) / 256, 1, NB * NC);
  warp_bicubic_kernel<<<grid, dim3(256, 1, 1), 0, stream>>>(img, sx, sy, sc, rot, out);
}